// SpatialBurgerDerivativeOperator_51273319580074
// MI455X (gfx1250) — compile-verified
//
#include <hip/hip_runtime.h>

#define N_NODES 1000000
#define N_EDGES 16000000

typedef int   v4i __attribute__((ext_vector_type(4)));
typedef float v4f __attribute__((ext_vector_type(4)));

// ---------------------------------------------------------------------------
// Edge scatter kernel, dense-u path (defined first so its asm is visible).
// 4 edges/thread. Edge stream read non-temporally (th:TH_LOAD_NT) so the
// 256 MB one-shot stream doesn't evict the hot 4 MB u / 4 MB out from L2.
// Division replaced by v_rcp_f32 + mul: the mask src*e > 0 guarantees e is
// finite and nonzero, so no IEEE special-case fixup is needed; ~1.5 ulp
// error is far below the tolerance implied by nondeterministic float-atomic
// accumulation order.
// ---------------------------------------------------------------------------
__global__ __launch_bounds__(256) void edge_kernel_dense(
    const int*   __restrict__ row,      // edge_index[0]
    const int*   __restrict__ col,      // edge_index[1]
    const float* __restrict__ eattr,    // (N_EDGES, 2) row-major
    const float* __restrict__ u,        // dense u[:,0] extract
    float*       __restrict__ out) {
  int i = blockIdx.x * blockDim.x + threadIdx.x;   // edge quad id

  v4i r4  = __builtin_nontemporal_load((const v4i*)row + i);
  v4i c4  = __builtin_nontemporal_load((const v4i*)col + i);
  v4f e01 = __builtin_nontemporal_load((const v4f*)eattr + 2 * i);
  v4f e23 = __builtin_nontemporal_load((const v4f*)eattr + 2 * i + 1);

  int   rr[4] = {r4.x, r4.y, r4.z, r4.w};
  int   cc[4] = {c4.x, c4.y, c4.z, c4.w};
  float ea[4] = {e01.x, e01.z, e23.x, e23.z};  // edge_attr[:,0] (stride 2)

#pragma unroll
  for (int k = 0; k < 4; ++k) {
    float src = u[rr[k]];
    float e   = ea[k];
    if (src * e > 0.0f) {
      // dest gather + atomic are EXEC-predicated under wave32: inactive lanes
      // (~50%) generate no cache traffic.
      float dst = u[cc[k]];
      float ld  = (dst - src) * __builtin_amdgcn_rcpf(e);
      __hip_atomic_fetch_add(out + cc[k], ld, __ATOMIC_RELAXED,
                             __HIP_MEMORY_SCOPE_AGENT);
    }
  }
}

// Fallback when workspace is too small for the dense-u extract: gather
// directly from strided x (x[n*4]).
__global__ __launch_bounds__(256) void edge_kernel_strided(
    const int*   __restrict__ row,
    const int*   __restrict__ col,
    const float* __restrict__ eattr,
    const float* __restrict__ x,
    float*       __restrict__ out) {
  int i = blockIdx.x * blockDim.x + threadIdx.x;

  v4i r4  = __builtin_nontemporal_load((const v4i*)row + i);
  v4i c4  = __builtin_nontemporal_load((const v4i*)col + i);
  v4f e01 = __builtin_nontemporal_load((const v4f*)eattr + 2 * i);
  v4f e23 = __builtin_nontemporal_load((const v4f*)eattr + 2 * i + 1);

  int   rr[4] = {r4.x, r4.y, r4.z, r4.w};
  int   cc[4] = {c4.x, c4.y, c4.z, c4.w};
  float ea[4] = {e01.x, e01.z, e23.x, e23.z};

#pragma unroll
  for (int k = 0; k < 4; ++k) {
    float src = x[(size_t)rr[k] * 4];
    float e   = ea[k];
    if (src * e > 0.0f) {
      float dst = x[(size_t)cc[k] * 4];
      float ld  = (dst - src) * __builtin_amdgcn_rcpf(e);
      __hip_atomic_fetch_add(out + cc[k], ld, __ATOMIC_RELAXED,
                             __HIP_MEMORY_SCOPE_AGENT);
    }
  }
}

// ---------------------------------------------------------------------------
// Fused prep: extract u = x[:,0] into a dense 4 MB array (L2-resident gather
// target) and zero the output accumulator. 4 nodes per thread. x reads are
// non-temporal; u / out stores use default RT policy so the lines stay hot
// in L2 for the edge kernel.
// ---------------------------------------------------------------------------
__global__ __launch_bounds__(256) void prep_kernel(const float* __restrict__ x,
                                                   float* __restrict__ u,
                                                   float* __restrict__ out,
                                                   int nquads) {
  int i = blockIdx.x * blockDim.x + threadIdx.x;   // node quad id
  if (i < nquads) {
    const v4f* xrow = (const v4f*)x + (size_t)i * 4;
    v4f r0 = __builtin_nontemporal_load(xrow + 0);
    v4f r1 = __builtin_nontemporal_load(xrow + 1);
    v4f r2 = __builtin_nontemporal_load(xrow + 2);
    v4f r3 = __builtin_nontemporal_load(xrow + 3);
    v4f uq = {r0.x, r1.x, r2.x, r3.x};
    v4f zq = {0.0f, 0.0f, 0.0f, 0.0f};
    ((v4f*)u)[i]   = uq;
    ((v4f*)out)[i] = zq;
  }
}

extern "C" void kernel_launch(void* const* d_in, const int* in_sizes, int n_in,
                              void* d_out, int out_size, void* d_ws, size_t ws_size,
                              hipStream_t stream) {
  const float* x     = (const float*)d_in[0];            // (N_NODES, 4) f32
  const int*   eidx  = (const int*)d_in[1];              // (2, N_EDGES) i32
  const float* eattr = (const float*)d_in[2];            // (N_EDGES, 2) f32
  const int*   row   = eidx;
  const int*   col   = eidx + N_EDGES;
  float*       out   = (float*)d_out;                    // (N_NODES,) f32

  const bool use_u = ws_size >= (size_t)N_NODES * sizeof(float);
  float* u = (float*)d_ws;

  // Prep: zero accumulator (+ extract dense u when workspace allows).
  {
    int nquads = N_NODES / 4;                 // 250000 (N_NODES % 4 == 0)
    int blocks = (nquads + 255) / 256;
    if (use_u) {
      prep_kernel<<<blocks, 256, 0, stream>>>(x, u, out, nquads);
    } else {
      prep_kernel<<<blocks, 256, 0, stream>>>(x, out, out, nquads);
    }
  }

  // Edge scatter: 4 edges per thread -> 4M threads -> 15625 blocks of 256.
  {
    int quads  = N_EDGES / 4;
    int blocks = quads / 256;   // 16M/4/256 = 15625 exactly
    if (use_u) {
      edge_kernel_dense<<<blocks, 256, 0, stream>>>(row, col, eattr, u, out);
    } else {
      edge_kernel_strided<<<blocks, 256, 0, stream>>>(row, col, eattr, x, out);
    }
  }
}